// Adaptive_Softmax_87522843560701
// MI455X (gfx1250) — compile-verified
//
#include <hip/hip_runtime.h>
#include <hip/hip_bf16.h>

#define HID   1024
#define VOC   50257
#define NCL   3
#define MROWS 32           // token rows per workgroup
#define NWAVE 8            // waves per workgroup (256 threads, wave32)

typedef __attribute__((ext_vector_type(2))) float v2f;
typedef __attribute__((ext_vector_type(8))) float v8f;

__device__ __constant__ int d_cuts[NCL + 1] = {0, 2000, 10000, VOC};

// online-softmax pairwise merge: (m,s) <- (m,s) + (om,os)
__device__ __forceinline__ void lse_merge(float& m, float& s, float om, float os) {
    float nm = fmaxf(m, om);
    s = s * __expf(m - nm) + os * __expf(om - nm);
    m = nm;
}

__global__ __launch_bounds__(256)
void Adaptive_Softmax_87522843560701_kernel(
        const float* __restrict__ x,        // [B, L+1, HID]
        const int*   __restrict__ y,        // [B*L]
        const float* __restrict__ cw,       // [HID, NCL]
        const float* __restrict__ cb,       // [NCL]
        const float* __restrict__ lw,       // [HID, VOC]
        const float* __restrict__ lb,       // [VOC]
        float*       __restrict__ out)      // [B*L]
{
    extern __shared__ float smem[];
    float* sx    = smem;                    // MROWS*HID   x-tile
    float* scl   = sx + MROWS * HID;        // MROWS*NCL   gate logits
    float* scomb = scl + MROWS * NCL;       // NWAVE*MROWS*3  (m,s,tgt)

    const int tid  = threadIdx.x;
    const int wid  = tid >> 5;
    const int lane = tid & 31;

    const int rBase = blockIdx.x * MROWS;         // first token row of this WG
    const int bIdx  = rBase >> 11;                // rows 0..2047 -> b=0, etc.
    const int l0    = rBase & 2047;
    const float* xg = x + ((size_t)bIdx * (2048 + 1) + l0) * HID;  // contiguous 32*HID chunk

    // ---- stage x tile to LDS (contiguous 128 KB) ----
    for (int i = tid; i < MROWS * HID / 4; i += 256) {
        float4 v = ((const float4*)xg)[i];
        ((float4*)sx)[i] = v;
    }
    __syncthreads();

    // ---- gate GEMM: 32 rows x 3 clusters, wave-parallel dot products ----
    for (int rr = 0; rr < MROWS / NWAVE; ++rr) {
        int row = wid * (MROWS / NWAVE) + rr;
        float a0 = 0.f, a1 = 0.f, a2 = 0.f;
        for (int j = 0; j < HID / 32; ++j) {
            int k = lane + j * 32;
            float xv = sx[row * HID + k];
            a0 += xv * cw[k * NCL + 0];
            a1 += xv * cw[k * NCL + 1];
            a2 += xv * cw[k * NCL + 2];
        }
        for (int d = 16; d; d >>= 1) {
            a0 += __shfl_xor(a0, d, 32);
            a1 += __shfl_xor(a1, d, 32);
            a2 += __shfl_xor(a2, d, 32);
        }
        if (lane == 0) {
            scl[row * NCL + 0] = a0 + cb[0];
            scl[row * NCL + 1] = a1 + cb[1];
            scl[row * NCL + 2] = a2 + cb[2];
        }
    }
    __syncthreads();

    // ---- per-lane slot bookkeeping (C-matrix layout: VGPR v -> M=v / v+8) ----
    const int colLocal = lane & 15;
    const int hi16     = lane >> 4;        // 0 or 1
    const int khalf    = hi16 << 1;        // A/B fragments: K offset 0 or 2

    int yslot[16];
#pragma unroll
    for (int t = 0; t < 2; ++t)
#pragma unroll
        for (int v = 0; v < 8; ++v) {
            int rl = t * 16 + v + (hi16 << 3);
            yslot[t * 8 + v] = y[rBase + rl];
        }

    const int aOff0 = colLocal * HID + khalf;          // tile0 (rows 0..15)
    const int aOff1 = (16 + colLocal) * HID + khalf;   // tile1 (rows 16..31)

    for (int ci = 0; ci < NCL; ++ci) {
        const int lcut = d_cuts[ci], rcut = d_cuts[ci + 1];

        float m[16], s[16], tg[16];
#pragma unroll
        for (int i = 0; i < 16; ++i) { m[i] = -1e30f; s[i] = 0.f; tg[i] = -1e30f; }

        // interleaved 16-wide column strips across the 8 waves
        for (int n0 = lcut + wid * 16; n0 < rcut; n0 += NWAVE * 16) {
            const int col  = n0 + colLocal;
            const int colc = min(col, VOC - 1);        // clamp ragged tail addressing
            const float* wp = lw + colc;

            v8f acc0 = {0.f, 0.f, 0.f, 0.f, 0.f, 0.f, 0.f, 0.f};
            v8f acc1 = {0.f, 0.f, 0.f, 0.f, 0.f, 0.f, 0.f, 0.f};

#pragma unroll 4
            for (int k = 0; k < HID; k += 4) {
                v2f a0 = *(const v2f*)(sx + aOff0 + k);    // ds_load_b64 (pairs fuse to 2addr)
                v2f a1 = *(const v2f*)(sx + aOff1 + k);
                int kk = k + khalf;
                v2f bf;
                bf.x = wp[kk * VOC];                       // B frag: K=kk,  N=colLocal
                bf.y = wp[(kk + 1) * VOC];                 //         K=kk+1
                acc0 = __builtin_amdgcn_wmma_f32_16x16x4_f32(
                           false, a0, false, bf, (short)0, acc0, false, false);
                acc1 = __builtin_amdgcn_wmma_f32_16x16x4_f32(
                           false, a1, false, bf, (short)0, acc1, false, false);
            }

            // branchless streaming segmented-softmax epilogue.
            // Out-of-range lanes feed -1e30: their m stays -1e30, so any spurious
            // s increments are annihilated (exp(m-nm)=0) at merge time.
            const float bias  = lb[colc];
            const bool  colIn = (col < rcut);
#pragma unroll
            for (int v = 0; v < 8; ++v) {
                float val0 = colIn ? acc0[v] + bias : -1e30f;
                float val1 = colIn ? acc1[v] + bias : -1e30f;
                { float nm = fmaxf(m[v], val0);
                  s[v] = s[v] * __expf(m[v] - nm) + __expf(val0 - nm);
                  m[v] = nm; }
                { float nm = fmaxf(m[8 + v], val1);
                  s[8 + v] = s[8 + v] * __expf(m[8 + v] - nm) + __expf(val1 - nm);
                  m[8 + v] = nm; }
                // col==yslot implies col<rcut (labels live inside their segment)
                tg[v]     = fmaxf(tg[v],     (col == yslot[v])     ? val0 : -1e30f);
                tg[8 + v] = fmaxf(tg[8 + v], (col == yslot[8 + v]) ? val1 : -1e30f);
            }
        }

        // ---- reduce across the 16 lanes sharing each row ----
#pragma unroll
        for (int sl = 0; sl < 16; ++sl) {
            float mm = m[sl], ss = s[sl], tt = tg[sl];
            for (int d = 1; d <= 8; d <<= 1) {
                float om = __shfl_xor(mm, d, 32);
                float os = __shfl_xor(ss, d, 32);
                float ot = __shfl_xor(tt, d, 32);
                lse_merge(mm, ss, om, os);
                tt = fmaxf(tt, ot);
            }
            m[sl] = mm; s[sl] = ss; tg[sl] = tt;
        }
        if (colLocal == 0) {
#pragma unroll
            for (int sl = 0; sl < 16; ++sl) {
                int rl  = (sl >> 3) * 16 + (sl & 7) + (hi16 << 3);
                int idx = (wid * MROWS + rl) * 3;
                scomb[idx + 0] = m[sl];
                scomb[idx + 1] = s[sl];
                scomb[idx + 2] = tg[sl];
            }
        }
        __syncthreads();

        // ---- combine waves, finalize rows whose label lives in this segment ----
        if (tid < MROWS) {
            int r = tid;
            float M = -1e30f, S = 0.f, T = -1e30f;
            for (int w = 0; w < NWAVE; ++w) {
                int idx = (w * MROWS + r) * 3;
                lse_merge(M, S, scomb[idx + 0], scomb[idx + 1]);
                T = fmaxf(T, scomb[idx + 2]);
            }
            int yv = y[rBase + r];
            if (yv >= lcut && yv < rcut) {
                float c0 = scl[r * NCL + 0], c1 = scl[r * NCL + 1], c2 = scl[r * NCL + 2];
                float cm  = fmaxf(c0, fmaxf(c1, c2));
                float cls = __logf(__expf(c0 - cm) + __expf(c1 - cm) + __expf(c2 - cm)) + cm;
                float cl_ll = scl[r * NCL + ci] - cls;
                float lse   = M + __logf(S);
                out[rBase + r] = -(cl_ll + T - lse);
            }
        }
        __syncthreads();
    }
}

extern "C" void kernel_launch(void* const* d_in, const int* in_sizes, int n_in,
                              void* d_out, int out_size, void* d_ws, size_t ws_size,
                              hipStream_t stream) {
    const float* x  = (const float*)d_in[0];
    const int*   y  = (const int*)  d_in[1];
    const float* cw = (const float*)d_in[2];
    const float* cb = (const float*)d_in[3];
    const float* lw = (const float*)d_in[4];
    const float* lb = (const float*)d_in[5];
    float* out = (float*)d_out;

    const int nTokens = out_size;                  // 4096
    const int nBlocks = nTokens / MROWS;           // 128 workgroups
    const size_t shmem = (size_t)(MROWS * HID + MROWS * NCL + NWAVE * MROWS * 3) * sizeof(float);

    Adaptive_Softmax_87522843560701_kernel<<<nBlocks, 256, shmem, stream>>>(
        x, y, cw, cb, lw, lb, out);
}